// KNNModule_58901181497623
// MI455X (gfx1250) — compile-verified
//
#include <hip/hip_runtime.h>
#include <hip/hip_bf16.h>

typedef float v2f __attribute__((ext_vector_type(2)));
typedef float v8f __attribute__((ext_vector_type(8)));

#define BATCH 2
#define N1    8192
#define N2    16384
#define DIM   64
#define TOPK  16
#define NTILES (N2 / 16)
#define NSPLIT 4
#define TPS    (NTILES / NSPLIT)   // tiles per split = 256

// ---------------------------------------------------------------------------
// Kernel 1: squared norms of every point row (p2 rows first, then p1 rows)
// ---------------------------------------------------------------------------
__global__ void knn_sqnorm_kernel(const float* __restrict__ p1,
                                  const float* __restrict__ p2,
                                  float* __restrict__ sq1,
                                  float* __restrict__ sq2) {
    int t = blockIdx.x * blockDim.x + threadIdx.x;
    const int total2 = BATCH * N2;
    const int total1 = BATCH * N1;
    if (t < total2) {
        const float4* row = (const float4*)(p2 + (size_t)t * DIM);
        float s = 0.f;
#pragma unroll
        for (int i = 0; i < DIM / 4; ++i) {
            float4 v = row[i];
            s += v.x * v.x + v.y * v.y + v.z * v.z + v.w * v.w;
        }
        sq2[t] = s;
    } else if (t < total2 + total1) {
        int r = t - total2;
        const float4* row = (const float4*)(p1 + (size_t)r * DIM);
        float s = 0.f;
#pragma unroll
        for (int i = 0; i < DIM / 4; ++i) {
            float4 v = row[i];
            s += v.x * v.x + v.y * v.y + v.z * v.z + v.w * v.w;
        }
        sq1[r] = s;
    }
}

// ---------------------------------------------------------------------------
// Kernel 2: fused WMMA distance tiles + per-row top-K over one N2 split.
// Single wave32 per workgroup. Cross-lane LDS exchange relies on lockstep +
// hardware in-order LDS processing; wave_barrier() only pins compiler order.
// ---------------------------------------------------------------------------
__global__ __launch_bounds__(32) void knn_topk_kernel(
        const float* __restrict__ p1, const float* __restrict__ p2,
        const float* __restrict__ sq2,
        float* __restrict__ part_d, int* __restrict__ part_i) {

    __shared__ float tile[16 * 18];      // stride 18 -> bank-conflict free
    __shared__ float sdl[16 * 32];       // two sorted 16-lists per row (dists)
    __shared__ int   sil[16 * 32];       // two sorted 16-lists per row (idx)

    const int lane = threadIdx.x;        // wave32
    const int half = lane >> 4;
    const int mr   = lane & 15;

    const int bid   = blockIdx.x;        // [0, BATCH*N1/16)
    const int b     = bid >> 9;          // N1/16 == 512
    const int qbase = (bid & 511) << 4;
    const int split = blockIdx.y;        // [0, NSPLIT)
    const int ntStart = split * TPS;

    const float* P1 = p1 + (size_t)b * N1 * DIM;
    const float* P2 = p2 + (size_t)b * N2 * DIM;
    const float* SQ2 = sq2 + (size_t)b * N2;

    // Loop-invariant A fragments (16x4 f32 layout: lanes = M, K pair by half).
    v2f afrag[16];
#pragma unroll
    for (int kk = 0; kk < 16; ++kk) {
        afrag[kk] = *(const v2f*)(P1 + (size_t)(qbase + mr) * DIM + 4 * kk + 2 * half);
    }

    float dlist[TOPK];
    int   ilist[TOPK];
#pragma unroll
    for (int t = 0; t < TOPK; ++t) { dlist[t] = 3.4e38f; ilist[t] = 0; }

    auto process = [&](int nbase, v2f (&bf)[16],
                       bool loadNext, int nbaseNext, v2f (&bn)[16]) {
        // D = A*B with inline-0 C; fold sq2 in at the epilogue.
        v8f acc = {};
#pragma unroll
        for (int kk = 0; kk < 16; ++kk) {
            acc = __builtin_amdgcn_wmma_f32_16x16x4_f32(
                false, afrag[kk], false, bf[kk], (short)0, acc, false, false);
        }

        if (loadNext) {
            const float* bpn = P2 + (size_t)(nbaseNext + mr) * DIM + 2 * half;
#pragma unroll
            for (int kk = 0; kk < 16; ++kk) bn[kk] = *(const v2f*)(bpn + 4 * kk);
        }

        const float s2 = SQ2[nbase + mr];   // column constant for this lane

        __builtin_amdgcn_wave_barrier();    // keep LDS stores after prior reads
#pragma unroll
        for (int r = 0; r < 8; ++r)
            tile[(r + 8 * half) * 18 + mr] = __builtin_fmaf(-2.0f, acc[r], s2);
        __builtin_amdgcn_wave_barrier();    // keep reads after stores

        // Gather the 8 candidate scores for this lane's row first.
        float vv[8];
#pragma unroll
        for (int j = 0; j < 8; ++j) vv[j] = tile[mr * 18 + half + 2 * j];
        __builtin_amdgcn_wave_barrier();

#pragma unroll
        for (int j = 0; j < 8; ++j) {
            const float v = vv[j];
            if (v < dlist[TOPK - 1]) {               // rare-taken guard
                dlist[TOPK - 1] = v;
                ilist[TOPK - 1] = nbase + half + 2 * j;
#pragma unroll
                for (int t = TOPK - 1; t >= 1; --t) {   // branch-free bubble
                    const bool  m  = dlist[t] < dlist[t - 1];
                    const float da = dlist[t - 1], db = dlist[t];
                    const int   ia = ilist[t - 1], ib = ilist[t];
                    dlist[t - 1] = m ? db : da;
                    dlist[t]     = m ? da : db;
                    ilist[t - 1] = m ? ib : ia;
                    ilist[t]     = m ? ia : ib;
                }
            }
        }
    };

    // Prime buffer A with the first tile of this split.
    v2f bufA[16], bufB[16];
    {
        const float* bp0 = P2 + (size_t)(ntStart * 16 + mr) * DIM + 2 * half;
#pragma unroll
        for (int kk = 0; kk < 16; ++kk) bufA[kk] = *(const v2f*)(bp0 + 4 * kk);
    }

    for (int i = 0; i < TPS; i += 2) {
        const int nb = (ntStart + i) * 16;
        process(nb,      bufA, true,          nb + 16, bufB);
        process(nb + 16, bufB, (i + 2) < TPS, nb + 32, bufA);
    }

    // Merge the two parity lists per row; write sorted partial list to ws.
    const int base = mr * 32 + half * 16;
    __builtin_amdgcn_wave_barrier();
#pragma unroll
    for (int t = 0; t < TOPK; ++t) { sdl[base + t] = dlist[t]; sil[base + t] = ilist[t]; }
    __builtin_amdgcn_wave_barrier();

    if (lane < 16) {
        const int m = lane;
        const size_t q = (size_t)b * N1 + qbase + m;
        const size_t obase = (q * NSPLIT + split) * TOPK;
        int i0 = 0, i1 = 0;
        for (int t = 0; t < TOPK; ++t) {
            const float v0 = sdl[m * 32 + (i0 < 16 ? i0 : 15)];
            const float v1 = sdl[m * 32 + 16 + (i1 < 16 ? i1 : 15)];
            const bool take0 = (i1 >= 16) || ((i0 < 16) && (v0 <= v1));
            part_d[obase + t] = take0 ? v0 : v1;
            part_i[obase + t] = take0 ? sil[m * 32 + i0] : sil[m * 32 + 16 + i1];
            i0 += take0 ? 1 : 0;
            i1 += take0 ? 0 : 1;
        }
    }
}

// ---------------------------------------------------------------------------
// Kernel 3: per-query NSPLIT-way merge of sorted partial lists; add ||p1||^2.
// ---------------------------------------------------------------------------
__global__ void knn_merge_kernel(const float* __restrict__ part_d,
                                 const int* __restrict__ part_i,
                                 const float* __restrict__ sq1,
                                 float* __restrict__ out_d,
                                 int* __restrict__ out_i) {
    const int q = blockIdx.x * blockDim.x + threadIdx.x;
    if (q >= BATCH * N1) return;
    const float* ld = part_d + (size_t)q * NSPLIT * TOPK;
    const int*   li = part_i + (size_t)q * NSPLIT * TOPK;
    const float  s1 = sq1[q];
    const size_t obase = (size_t)q * TOPK;

    int p0 = 0, p1 = 0, p2 = 0, p3 = 0;
    for (int t = 0; t < TOPK; ++t) {
        const float v0 = (p0 < TOPK) ? ld[0 * TOPK + p0] : 3.5e38f;
        const float v1 = (p1 < TOPK) ? ld[1 * TOPK + p1] : 3.5e38f;
        const float v2 = (p2 < TOPK) ? ld[2 * TOPK + p2] : 3.5e38f;
        const float v3 = (p3 < TOPK) ? ld[3 * TOPK + p3] : 3.5e38f;
        float best = v0; int bs = 0;
        if (v1 < best) { best = v1; bs = 1; }
        if (v2 < best) { best = v2; bs = 2; }
        if (v3 < best) { best = v3; bs = 3; }
        const int bp = (bs == 0) ? p0 : (bs == 1) ? p1 : (bs == 2) ? p2 : p3;
        out_d[obase + t] = best + s1;
        out_i[obase + t] = li[bs * TOPK + bp];
        p0 += (bs == 0); p1 += (bs == 1); p2 += (bs == 2); p3 += (bs == 3);
    }
}

// ---------------------------------------------------------------------------
extern "C" void kernel_launch(void* const* d_in, const int* in_sizes, int n_in,
                              void* d_out, int out_size, void* d_ws, size_t ws_size,
                              hipStream_t stream) {
    const float* p1 = (const float*)d_in[0];   // [2, 8192, 64]
    const float* p2 = (const float*)d_in[1];   // [2, 16384, 64]

    float* out_d = (float*)d_out;
    int*   out_i = (int*)d_out + (size_t)BATCH * N1 * TOPK;

    // Workspace layout: sq2 | sq1 | partial dists | partial idx
    float* sq2    = (float*)d_ws;                        // BATCH*N2
    float* sq1    = sq2 + (size_t)BATCH * N2;            // BATCH*N1
    float* part_d = sq1 + (size_t)BATCH * N1;            // BATCH*N1*NSPLIT*TOPK
    int*   part_i = (int*)(part_d + (size_t)BATCH * N1 * NSPLIT * TOPK);

    const int total = BATCH * (N1 + N2);
    knn_sqnorm_kernel<<<(total + 255) / 256, 256, 0, stream>>>(p1, p2, sq1, sq2);

    dim3 grid(BATCH * (N1 / 16), NSPLIT);    // 1024 x 4 waves
    knn_topk_kernel<<<grid, 32, 0, stream>>>(p1, p2, sq2, part_d, part_i);

    const int nq = BATCH * N1;
    knn_merge_kernel<<<(nq + 255) / 256, 256, 0, stream>>>(part_d, part_i, sq1,
                                                           out_d, out_i);
}